// VectorQuantizer_72215580115687
// MI455X (gfx1250) — compile-verified
//
#include <hip/hip_runtime.h>
#include <math.h>

typedef float v2f __attribute__((ext_vector_type(2)));
typedef float v4f __attribute__((ext_vector_type(4)));
typedef float v8f __attribute__((ext_vector_type(8)));

#define NUM_EMB 1024
#define DIM     256
#define N_TOT   65536
#define HW      1024
#define ROWS    128
#define ASTR    260   // padded LDS stride (floats): 4*r mod 64 -> conflict-free

// d_out layout (floats):
//   [0]                          loss
//   [1 .. 16777216]              quantized_out  (NCHW, 64x256x32x32)
//   [16777217]                   perplexity
//   [16777218 .. +67108863]      encodings      (65536 x 1024)
//   [83886082 .. +16777215]      quantized_flat (65536 x 256)
#define OFF_QOUT 1
#define OFF_PERP 16777217
#define OFF_ENC  16777218
#define OFF_QF   83886082

// ws layout: [0..4095] float esq[1024]; [4096..8191] int counts[1024];
//            [8192..10239] float lossPartial[512]

// ---------------- zero-fill encodings (268 MB of zeros, b64 stores) ----------
__global__ void vq_zero_enc(v2f* __restrict__ p, long n) {
  long i = (long)blockIdx.x * blockDim.x + threadIdx.x;
  long stride = (long)gridDim.x * blockDim.x;
  v2f z = {0.f, 0.f};
  for (; i < n; i += stride) p[i] = z;
}

// ---------------- e_sq[k] = sum_d emb[k][d]^2 ; also zero counts -------------
__global__ void vq_prep(const float* __restrict__ emb,
                        float* __restrict__ esq, int* __restrict__ counts) {
  int k = blockIdx.x * 256 + threadIdx.x;    // 4 blocks x 256 = 1024
  counts[k] = 0;
  const v4f* row = (const v4f*)(emb + (size_t)k * DIM);
  float s = 0.f;
#pragma unroll 4
  for (int j = 0; j < DIM / 4; ++j) {
    v4f v = row[j];
    s += v.x * v.x + v.y * v.y + v.z * v.z + v.w * v.w;
  }
  esq[k] = s;
}

// ---------------- main: distances via WMMA, argmin, outputs ------------------
__global__ __launch_bounds__(256, 1) void vq_main(
    const float* __restrict__ x,      // inputs NCHW [64,256,32,32]
    const float* __restrict__ emb,    // [1024,256]
    const float* __restrict__ esq_g,  // [1024]
    int*   __restrict__ counts,
    float* __restrict__ lossP,        // [512] per-block partials
    float* __restrict__ qout,         // d_out + OFF_QOUT (NCHW)
    float* __restrict__ enc,          // d_out + OFF_ENC
    float* __restrict__ qflat)        // d_out + OFF_QF
{
  __shared__ float A[ROWS * ASTR];     // 133,120 B : x tile, [row][d]
  __shared__ float Bt[16 * ASTR];      //  16,640 B : codebook tile, [k][d]
  __shared__ float esq[NUM_EMB];       //   4,096 B
  __shared__ int   idxs[ROWS];         //     512 B
  __shared__ float red[256];           //   1,024 B

  const int tid  = threadIdx.x;
  const int lane = tid & 31;
  const int wv   = tid >> 5;      // 8 waves, 16 rows each
  const int half = lane >> 4;
  const int lm   = lane & 15;

  const int n0  = blockIdx.x * ROWS;
  const int b   = n0 >> 10;          // 1024 rows per batch image; 128 | 1024
  const int hw0 = n0 & 1023;
  const float* xb = x + (size_t)b * (DIM * HW) + hw0;

  // Stage A tile: A[r][d] = inputs[b, d, hw0+r]  (coalesced over r)
  for (int i = tid; i < ROWS * DIM; i += 256) {
    int r = i & (ROWS - 1);
    int d = i >> 7;
    A[r * ASTR + d] = xb[(size_t)d * HW + r];
  }
  for (int j = tid; j < NUM_EMB; j += 256) esq[j] = esq_g[j];
  __syncthreads();

  // Hold the wave's full A operand (16 rows x 256 d) in registers:
  // 16x4 f32 A-fragment: lane holds row M=lm, d = d0 + 2*half + {0,1}
  const float* aptr = &A[(wv * 16 + lm) * ASTR + 2 * half];
  v2f afrag[64];
#pragma unroll
  for (int t = 0; t < 64; ++t) afrag[t] = *(const v2f*)(aptr + 4 * t);

  float minv[8];
  int   mini[8];
#pragma unroll
  for (int j = 0; j < 8; ++j) { minv[j] = 3.0e38f; mini[j] = 0; }

  for (int k0 = 0; k0 < NUM_EMB; k0 += 16) {
    // Stage 16 codebook rows into LDS (coalesced over d)
    for (int i = tid; i < 16 * DIM; i += 256) {
      int d  = i & (DIM - 1);
      int kr = i >> 8;
      Bt[kr * ASTR + d] = emb[(size_t)(k0 + kr) * DIM + d];
    }
    // prefetch next tile into caches while we compute (global_prefetch_b8)
    if (k0 + 16 < NUM_EMB)
      __builtin_prefetch(emb + (size_t)(k0 + 16) * DIM + tid * 16, 0, 3);
    __syncthreads();

    // 16x16 dot tile over D=256: 64 x V_WMMA_F32_16X16X4_F32
    v8f acc = {0.f, 0.f, 0.f, 0.f, 0.f, 0.f, 0.f, 0.f};
    const float* bptr = &Bt[lm * ASTR + 2 * half];  // B frag: N=lm, K=2h+{0,1}
#pragma unroll
    for (int t = 0; t < 64; ++t) {
      v2f bfrag = *(const v2f*)(bptr + 4 * t);
      acc = __builtin_amdgcn_wmma_f32_16x16x4_f32(
          false, afrag[t], false, bfrag, (short)0, acc, false, false);
    }

    // score = ||e||^2 - 2*dot ; running argmin (ascending k keeps first-min)
    float es   = esq[k0 + lm];
    int   kcol = k0 + lm;
#pragma unroll
    for (int j = 0; j < 8; ++j) {
      float sc = fmaf(-2.0f, acc[j], es);
      if (sc < minv[j]) { minv[j] = sc; mini[j] = kcol; }
    }
    __syncthreads();   // protect Bt before next refill
  }

  // Cross-lane reduce within each 16-lane half (rows j / 8+j), tie -> low idx
#pragma unroll
  for (int j = 0; j < 8; ++j) {
#pragma unroll
    for (int mask = 8; mask >= 1; mask >>= 1) {
      float ov = __shfl_xor(minv[j], mask, 32);
      int   oi = __shfl_xor(mini[j], mask, 32);
      if (ov < minv[j] || (ov == minv[j] && oi < mini[j])) {
        minv[j] = ov; mini[j] = oi;
      }
    }
  }
  if (lm == 0) {
#pragma unroll
    for (int j = 0; j < 8; ++j) idxs[wv * 16 + half * 8 + j] = mini[j];
  }
  __syncthreads();

  // Encodings scatter + histogram (integer atomics: deterministic)
  if (tid < ROWS) {
    int k = idxs[tid];
    atomicAdd(&counts[k], 1);
    enc[(size_t)(n0 + tid) * NUM_EMB + k] = 1.0f;
  }

  // quantized (straight-through) in NCHW + loss partial; emb rows are L2-hot
  float lsum = 0.f;
  for (int i = tid; i < ROWS * DIM; i += 256) {
    int r = i & (ROWS - 1);
    int d = i >> 7;
    float q    = emb[(size_t)idxs[r] * DIM + d];
    float xv   = A[r * ASTR + d];
    float diff = q - xv;
    lsum += diff * diff;
    qout[(size_t)b * (DIM * HW) + (size_t)d * HW + hw0 + r] = xv + diff;
  }
  red[tid] = lsum;
  __syncthreads();
  for (int s = 128; s > 0; s >>= 1) {
    if (tid < s) red[tid] += red[tid + s];
    __syncthreads();
  }
  if (tid == 0) lossP[blockIdx.x] = red[0];

  // quantized_flat, coalesced over d
  for (int i = tid; i < ROWS * DIM; i += 256) {
    int d = i & (DIM - 1);
    int r = i >> 8;
    qflat[(size_t)(n0 + r) * DIM + d] = emb[(size_t)idxs[r] * DIM + d];
  }
}

// ---------------- finalize: loss + perplexity --------------------------------
__global__ void vq_final(const int* __restrict__ counts,
                         const float* __restrict__ lossP,
                         float* __restrict__ out_loss,
                         float* __restrict__ out_perp) {
  __shared__ float red[256];
  int tid = threadIdx.x;
  red[tid] = lossP[tid] + lossP[tid + 256];
  __syncthreads();
  for (int s = 128; s > 0; s >>= 1) {
    if (tid < s) red[tid] += red[tid + s];
    __syncthreads();
  }
  if (tid == 0) {
    float m = red[0] * (1.0f / ((float)N_TOT * (float)DIM));
    *out_loss = m + 0.25f * m;   // q_latent + commitment*e_latent (identical means)
  }
  __syncthreads();
  float e = 0.f;
  for (int k = tid; k < NUM_EMB; k += 256) {
    float p = (float)counts[k] * (1.0f / (float)N_TOT);
    e += p * logf(p + 1e-10f);
  }
  red[tid] = e;
  __syncthreads();
  for (int s = 128; s > 0; s >>= 1) {
    if (tid < s) red[tid] += red[tid + s];
    __syncthreads();
  }
  if (tid == 0) *out_perp = expf(-red[0]);
}

extern "C" void kernel_launch(void* const* d_in, const int* in_sizes, int n_in,
                              void* d_out, int out_size, void* d_ws, size_t ws_size,
                              hipStream_t stream) {
  const float* x   = (const float*)d_in[0];   // [64,256,32,32] f32
  const float* emb = (const float*)d_in[1];   // [1024,256] f32
  float* out = (float*)d_out;

  float* esq    = (float*)d_ws;
  int*   counts = (int*)((char*)d_ws + 4096);
  float* lossP  = (float*)((char*)d_ws + 8192);

  float* out_loss = out;
  float* qout     = out + (size_t)OFF_QOUT;
  float* out_perp = out + (size_t)OFF_PERP;
  float* enc      = out + (size_t)OFF_ENC;   // 8-byte aligned base
  float* qflat    = out + (size_t)OFF_QF;

  long nf2 = 67108864L / 2;  // encodings as float2
  vq_zero_enc<<<16384, 256, 0, stream>>>((v2f*)enc, nf2);
  vq_prep<<<4, 256, 0, stream>>>(emb, esq, counts);
  vq_main<<<512, 256, 0, stream>>>(x, emb, esq, counts, lossP, qout, enc, qflat);
  vq_final<<<1, 256, 0, stream>>>(counts, lossP, out_loss, out_perp);
}